// Vec2Gram_54855322305115
// MI455X (gfx1250) — compile-verified
//
#include <hip/hip_runtime.h>

// gram[n,c,e] = sum_d u[c,d] * v[n,d] * u[e,d]
// N=256, D=4096, C=64.  Per-n GEMM: A = u row-scaled by v[n] [64x4096], B = u^T.
// One workgroup (4 waves) per n. Each wave computes a 16x64 strip of the 64x64
// gram: the A fragment (and its v-scaling) is loaded once per k-step and reused
// across 4 chained V_WMMA_F32_16X16X4_F32 ops -> 6 b64 LDS loads per 4 WMMAs.

typedef __attribute__((ext_vector_type(2))) float v2f;
typedef __attribute__((ext_vector_type(8))) float v8f;

#define N_   256
#define D_   4096
#define C_   64
#define KC   128         // K-chunk staged in LDS per iteration
#define LDST 132         // padded LDS row stride (dwords): %64 != 0 (bank spread),
                         // %4 == 0 (16B-aligned float4 staging stores)

__global__ __launch_bounds__(128)
void vec2gram_wmma_f32(const float* __restrict__ v,
                       const float* __restrict__ u,
                       float* __restrict__ out) {
    __shared__ float uS[C_ * LDST];   // u[0:64, k0:k0+128], padded rows
    __shared__ float vS[KC];          // v[n, k0:k0+128]

    const int n    = blockIdx.x;
    const int tid  = threadIdx.x;     // 0..127
    const int wave = tid >> 5;        // 0..3  -> output row-tile ct
    const int lane = tid & 31;
    const int half = lane >> 4;       // 0: K={0,1}, 1: K={2,3} within a k-step
    const int col  = lane & 15;       // A: row M, B: col N

    const float* __restrict__ vrow = v + n * D_;

    v8f acc[4] = {};                  // 4 et-tiles of the 16x64 strip

    for (int k0 = 0; k0 < D_; k0 += KC) {
        __syncthreads();              // previous chunk's LDS reads done

        // ---- cooperative staging: u[64][128] = 2048 float4 / 128 threads ----
        #pragma unroll
        for (int i = 0; i < 16; ++i) {
            const int f   = tid + i * 128;     // float4 index 0..2047
            const int row = f >> 5;            // 0..63
            const int c4  = (f & 31) << 2;     // 0,4,...,124
            const float4 g =
                *reinterpret_cast<const float4*>(u + row * D_ + k0 + c4);
            *reinterpret_cast<float4*>(&uS[row * LDST + c4]) = g;
        }
        if (tid < 32) {
            const float4 g =
                *reinterpret_cast<const float4*>(vrow + k0 + (tid << 2));
            *reinterpret_cast<float4*>(&vS[tid << 2]) = g;
        }
        __syncthreads();

        // ---- 32 k-steps, 4 WMMAs each (shared scaled-A fragment) ----
        const float* aBase = &uS[(wave * 16 + col) * LDST + half * 2];
        const float* bBase = &uS[col * LDST + half * 2];   // et=0 row block
        const float* vBase = &vS[half * 2];

        #pragma unroll
        for (int kk = 0; kk < KC; kk += 4) {
            const v2f au = *reinterpret_cast<const v2f*>(aBase + kk);
            const v2f vv = *reinterpret_cast<const v2f*>(vBase + kk);
            const v2f a  = au * vv;                        // row-scaled A frag

            #pragma unroll
            for (int t = 0; t < 4; ++t) {                  // et tiles 0..3
                const v2f bu = *reinterpret_cast<const v2f*>(
                    bBase + t * (16 * LDST) + kk);         // B = u^T fragment
                acc[t] = __builtin_amdgcn_wmma_f32_16x16x4_f32(
                             /*neg_a=*/false, a, /*neg_b=*/false, bu,
                             /*c_mod=*/(short)0, acc[t],
                             /*reuse_a=*/false, /*reuse_b=*/false);
            }
        }
    }

    // ---- epilogue: C/D layout -> gram[n, wave*16 + half*8 + r, et*16 + col] ----
    #pragma unroll
    for (int t = 0; t < 4; ++t) {
        float* obase = out + n * (C_ * C_)
                     + (wave * 16 + half * 8) * C_ + t * 16 + col;
        #pragma unroll
        for (int r = 0; r < 8; ++r) {
            obase[r * C_] = acc[t][r];
        }
    }
}

extern "C" void kernel_launch(void* const* d_in, const int* in_sizes, int n_in,
                              void* d_out, int out_size, void* d_ws, size_t ws_size,
                              hipStream_t stream) {
    const float* v = (const float*)d_in[0];   // [256, 4096] fp32
    const float* u = (const float*)d_in[1];   // [64, 4096]  fp32
    float* out     = (float*)d_out;           // [256, 64, 64] fp32

    vec2gram_wmma_f32<<<dim3(N_), dim3(128), 0, stream>>>(v, u, out);
}